// LightGCN_Encoder_24953759990566
// MI455X (gfx1250) — compile-verified
//
#include <hip/hip_runtime.h>

#ifndef __has_builtin
#define __has_builtin(x) 0
#endif

#define BLOCK 256            // 8 waves (wave32)
#define EPC   BLOCK          // edges per block chunk (1 edge per thread slot)

// ---------------- CDNA5 async global->LDS helpers ----------------
#if defined(__AMDGCN__) && __has_builtin(__builtin_amdgcn_global_load_async_to_lds_b32)
#define HAVE_ASYNC_LDS 1
typedef __attribute__((address_space(1))) int g_int;   // global address space
typedef __attribute__((address_space(3))) int l_int;   // LDS address space
__device__ __forceinline__ void async_g2l_b32(const void* g, void* l) {
    __builtin_amdgcn_global_load_async_to_lds_b32((g_int*)g, (l_int*)l, 0, 0);
}
__device__ __forceinline__ void async_wait0() {
#if __has_builtin(__builtin_amdgcn_s_wait_asynccnt)
    __builtin_amdgcn_s_wait_asynccnt(0);
#else
    asm volatile("s_wait_asynccnt 0" ::: "memory");
#endif
}
__device__ __forceinline__ void async_wait3() {
#if __has_builtin(__builtin_amdgcn_s_wait_asynccnt)
    __builtin_amdgcn_s_wait_asynccnt(3);
#else
    asm volatile("s_wait_asynccnt 3" ::: "memory");
#endif
}
#else
#define HAVE_ASYNC_LDS 0
__device__ __forceinline__ void async_wait0() {}
__device__ __forceinline__ void async_wait3() {}
#endif

__device__ __forceinline__ void atomic_add_f32(float* p, float v) {
    // Direct global_atomic_add_f32 (no return, no CAS loop)
    unsafeAtomicAdd(p, v);
}

// ---------------- init: A = concat(user,item); B = acc = 0 ----------------
__global__ void init_kernel(const float4* __restrict__ ue, const float4* __restrict__ ie,
                            int uN4, int n4,
                            float4* __restrict__ A, float4* __restrict__ B,
                            float4* __restrict__ acc) {
    int i = blockIdx.x * blockDim.x + threadIdx.x;
    if (i < n4) {
        A[i] = (i < uN4) ? ue[i] : ie[i - uN4];
        B[i]   = make_float4(0.f, 0.f, 0.f, 0.f);
        acc[i] = make_float4(0.f, 0.f, 0.f, 0.f);
    }
}

// ---------------- SpMM: y[r,:] += v * x[c,:] (D = 64) ----------------
// One wave processes one edge at a time: 32 lanes x float2 = 64 floats.
// Wave w stages and consumes ONLY LDS slots [32w, 32w+32) => no barriers.
// Edge triples are double-buffered via async global->LDS + ASYNCcnt:
// next chunk's 3 async loads fly while the current 32 edges are processed.
__global__ void __launch_bounds__(BLOCK)
spmm_kernel(const int* __restrict__ row, const int* __restrict__ col,
            const float* __restrict__ val,
            const float* __restrict__ x, float* __restrict__ y, int nnz) {
    __shared__ int   s_row[2][EPC];
    __shared__ int   s_col[2][EPC];
    __shared__ float s_val[2][EPC];

    const int lane = threadIdx.x & 31;
    const int wave = threadIdx.x >> 5;      // 0..7
    const int t    = threadIdx.x;
    const int nchunks = (nnz + EPC - 1) / EPC;
    const int lo   = wave * 32;

    // stage first chunk into buffer 0 (clamped index keeps EXEC uniform)
    {
        int e  = blockIdx.x * EPC + t;
        int ee = e < nnz ? e : nnz - 1;
#if HAVE_ASYNC_LDS
        async_g2l_b32(&row[ee], &s_row[0][t]);
        async_g2l_b32(&col[ee], &s_col[0][t]);
        async_g2l_b32(&val[ee], &s_val[0][t]);
#else
        s_row[0][t] = row[ee];
        s_col[0][t] = col[ee];
        s_val[0][t] = val[ee];
#endif
    }

    int cur = 0;
    for (int ch = blockIdx.x; ch < nchunks; ch += gridDim.x) {
        const int next = ch + gridDim.x;
        if (next < nchunks) {
            // prefetch next chunk into the alternate buffer
            int e  = next * EPC + t;
            int ee = e < nnz ? e : nnz - 1;
#if HAVE_ASYNC_LDS
            async_g2l_b32(&row[ee], &s_row[cur ^ 1][t]);
            async_g2l_b32(&col[ee], &s_col[cur ^ 1][t]);
            async_g2l_b32(&val[ee], &s_val[cur ^ 1][t]);
            async_wait3();   // oldest 3 (current buffer) retired; new 3 in flight
#else
            s_row[cur ^ 1][t] = row[ee];
            s_col[cur ^ 1][t] = col[ee];
            s_val[cur ^ 1][t] = val[ee];
#endif
        } else {
            async_wait0();   // drain: current buffer ready
        }

        const int base = ch * EPC;
        const int cnt  = (nnz - base < EPC) ? (nnz - base) : EPC;
        if (cnt == EPC) {
            // fast path: full chunk, unrolled so several gathers are in flight
            #pragma unroll 4
            for (int k = 0; k < 32; ++k) {
                const int   r = s_row[cur][lo + k];   // LDS broadcast reads
                const int   c = s_col[cur][lo + k];
                const float v = s_val[cur][lo + k];
                const float2 xv = *(const float2*)(x + (size_t)c * 64 + lane * 2);
                float* yp = y + (size_t)r * 64 + lane * 2;
                atomic_add_f32(yp,     v * xv.x);
                atomic_add_f32(yp + 1, v * xv.y);
            }
        } else {
            const int hi = (lo + 32 < cnt) ? (lo + 32) : cnt;
            for (int k = lo; k < hi; ++k) {
                const int   r = s_row[cur][k];
                const int   c = s_col[cur][k];
                const float v = s_val[cur][k];
                const float2 xv = *(const float2*)(x + (size_t)c * 64 + lane * 2);
                float* yp = y + (size_t)r * 64 + lane * 2;
                atomic_add_f32(yp,     v * xv.x);
                atomic_add_f32(yp + 1, v * xv.y);
            }
        }
        cur ^= 1;
    }
}

// ---------------- acc += y ; xold = 0 (prep for next layer) ----------------
__global__ void accum_clear_kernel(float4* __restrict__ acc, const float4* __restrict__ y,
                                   float4* __restrict__ xold, int n4) {
    int i = blockIdx.x * blockDim.x + threadIdx.x;
    if (i < n4) {
        float4 a = acc[i], b = y[i];
        a.x += b.x; a.y += b.y; a.z += b.z; a.w += b.w;
        acc[i]  = a;
        xold[i] = make_float4(0.f, 0.f, 0.f, 0.f);
    }
}

// ---------------- out = (acc + y_last) / 3 ----------------
__global__ void final_kernel(const float4* __restrict__ acc, const float4* __restrict__ y,
                             float4* __restrict__ out, int n4) {
    int i = blockIdx.x * blockDim.x + threadIdx.x;
    if (i < n4) {
        const float s = 1.0f / 3.0f;
        float4 a = acc[i], b = y[i];
        out[i] = make_float4((a.x + b.x) * s, (a.y + b.y) * s,
                             (a.z + b.z) * s, (a.w + b.w) * s);
    }
}

extern "C" void kernel_launch(void* const* d_in, const int* in_sizes, int n_in,
                              void* d_out, int out_size, void* d_ws, size_t ws_size,
                              hipStream_t stream) {
    (void)n_in; (void)out_size; (void)ws_size;
    const float* ue  = (const float*)d_in[0];   // user_emb (U*64)
    const float* ie  = (const float*)d_in[1];   // item_emb (I*64)
    const float* val = (const float*)d_in[2];   // edge_val (NNZ)
    const int*   row = (const int*)  d_in[3];   // edge_row
    const int*   col = (const int*)  d_in[4];   // edge_col

    const int uN    = in_sizes[0];
    const int iN    = in_sizes[1];
    const int nnz   = in_sizes[2];
    const int total = uN + iN;                  // N * 64 floats

    float* A   = (float*)d_ws;
    float* B   = A + (size_t)total;
    float* acc = B + (size_t)total;
    float* out = (float*)d_out;

    const int n4 = total / 4;
    const int gb = (n4 + 255) / 256;

    const int nchunks = (nnz + EPC - 1) / EPC;
    int gs = nchunks < 4096 ? nchunks : 4096;

    init_kernel<<<gb, 256, 0, stream>>>((const float4*)ue, (const float4*)ie,
                                        uN / 4, n4, (float4*)A, (float4*)B, (float4*)acc);
    // layer 1: A -> B
    spmm_kernel<<<gs, BLOCK, 0, stream>>>(row, col, val, A, B, nnz);
    accum_clear_kernel<<<gb, 256, 0, stream>>>((float4*)acc, (const float4*)B, (float4*)A, n4);
    // layer 2: B -> A
    spmm_kernel<<<gs, BLOCK, 0, stream>>>(row, col, val, B, A, nnz);
    accum_clear_kernel<<<gb, 256, 0, stream>>>((float4*)acc, (const float4*)A, (float4*)B, n4);
    // layer 3: A -> B
    spmm_kernel<<<gs, BLOCK, 0, stream>>>(row, col, val, A, B, nnz);
    final_kernel<<<gb, 256, 0, stream>>>((const float4*)acc, (const float4*)B, (float4*)out, n4);
}